// BlockwiseSelector_20005957665573
// MI455X (gfx1250) — compile-verified
//
#include <hip/hip_runtime.h>
#include <hip/hip_bf16.h>
#include <math.h>

// ---------------------------------------------------------------------------
// Problem constants (from reference)
// ---------------------------------------------------------------------------
#define B_DIM        4
#define H_DIM        32
#define G_DIM        4      // NUM_QUERY_GROUPS
#define HPG          8      // heads per group = H/G
#define NCK          128    // compressed keys per head
#define D_DIM        128    // head dim
#define SEQ          8192
#define BLK          64     // BLOCK_SIZE
#define NSEL         16     // NUM_SELECTED
#define SEL_ROWS     (NSEL * BLK)              // 1024 rows per (b,g)
#define KEY_ELEMS    ((size_t)B_DIM * G_DIM * SEL_ROWS * D_DIM) // 2,097,152

typedef __attribute__((ext_vector_type(2))) float v2f;
typedef __attribute__((ext_vector_type(8))) float v8f;

// ---------------------------------------------------------------------------
// Kernel 1: scores via V_WMMA_F32_16X16X4_F32, softmax, group-mean, top-16.
// One workgroup per (b,g); one wave32 per head (8 waves).
// Writes 16 selected compressed-block indices (in top-k rank order) per (b,g)
// into d_ws as int32.
// ---------------------------------------------------------------------------
__global__ __launch_bounds__(256) void nsa_select_kernel(
    const float* __restrict__ query,           // (B,H,1,D)
    const float* __restrict__ ck,              // (B,H,NCK,D)
    int* __restrict__ sel)                     // (B*G*NSEL)
{
    __shared__ float s_probs[HPG][NCK];  // per-head softmax probabilities
    __shared__ float s_avg[NCK];         // group-averaged probabilities

    const int bg   = blockIdx.x;         // 0..15
    const int b    = bg >> 2;
    const int g    = bg & 3;
    const int wave = threadIdx.x >> 5;   // head within group, 0..7
    const int lane = threadIdx.x & 31;
    const int l15  = lane & 15;
    const int half = lane >> 4;          // 0: K=0,1  1: K=2,3  (A layout halves)
    const int h    = g * HPG + wave;

    const float* qh  = query + ((size_t)(b * H_DIM + h)) * D_DIM;
    const float* ckh = ck    + ((size_t)(b * H_DIM + h)) * NCK * D_DIM;

    const float scale = 0.08838834764831845f; // 1/sqrt(128)

    // ---- scores[n] = sum_d ck[n,d] * q[d], 16 rows per WMMA tile ----
    // A: 16x4 f32 tile of ck rows (lane l15 = row, half selects K pair).
    // B: 4x16 f32 tile with every column equal to the q K-chunk, so the
    //    result D[i][j] = scores[n0+i] replicated across columns.
    for (int n0 = 0; n0 < NCK; n0 += 16) {
        v8f acc = {};
        const float* arow = ckh + (size_t)(n0 + l15) * D_DIM;
        #pragma unroll 4
        for (int d0 = 0; d0 < D_DIM; d0 += 4) {
            const int kb = d0 + 2 * half;            // even -> 8B aligned
            v2f a  = *(const v2f*)(arow + kb);       // A: ck[n, kb], ck[n, kb+1]
            v2f bb = *(const v2f*)(qh + kb);         // B: q[kb], q[kb+1] (all cols)
            acc = __builtin_amdgcn_wmma_f32_16x16x4_f32(
                /*neg_a=*/false, a, /*neg_b=*/false, bb,
                /*c_mod=*/(short)0, acc, /*reuse_a=*/false, /*reuse_b=*/false);
        }
        // C/D layout: lanes 0-15 hold M=0..7 in acc[0..7]; lanes 16-31 M=8..15.
        // Columns are replicated -> let lane 0 and lane 16 write the 16 scores.
        if (l15 == 0) {
            float* srow = &s_probs[wave][n0 + half * 8];
            #pragma unroll
            for (int v = 0; v < 8; ++v) srow[v] = acc[v];
        }
    }
    __syncthreads();

    // ---- softmax over n (128) per head, wave-local: 4 values per lane ----
    {
        float vals[4];
        float vmax = -INFINITY;
        #pragma unroll
        for (int i = 0; i < 4; ++i) {
            vals[i] = s_probs[wave][lane * 4 + i] * scale;
            vmax = fmaxf(vmax, vals[i]);
        }
        #pragma unroll
        for (int off = 16; off > 0; off >>= 1)
            vmax = fmaxf(vmax, __shfl_xor(vmax, off, 32));
        float vsum = 0.f;
        #pragma unroll
        for (int i = 0; i < 4; ++i) {
            vals[i] = expf(vals[i] - vmax);
            vsum += vals[i];
        }
        #pragma unroll
        for (int off = 16; off > 0; off >>= 1)
            vsum += __shfl_xor(vsum, off, 32);
        const float inv = 1.0f / vsum;
        #pragma unroll
        for (int i = 0; i < 4; ++i)
            s_probs[wave][lane * 4 + i] = vals[i] * inv;
    }
    __syncthreads();

    // ---- mean over the 8 heads of this group ----
    const int t = threadIdx.x;
    if (t < NCK) {
        float acc = 0.f;
        #pragma unroll
        for (int w = 0; w < HPG; ++w) acc += s_probs[w][t];
        s_avg[t] = acc * (1.0f / HPG);
    }
    __syncthreads();

    // ---- stable top-16 by rank counting (matches jax.lax.top_k order) ----
    if (t < NCK) {
        const float p = s_avg[t];
        int rank = 0;
        #pragma unroll 8
        for (int m = 0; m < NCK; ++m) {
            const float pm = s_avg[m];
            rank += (pm > p) || (pm == p && m < t);
        }
        if (rank < NSEL) sel[bg * NSEL + rank] = t;
    }
}

// ---------------------------------------------------------------------------
// Kernel 2: bandwidth-bound gather of selected 64-row blocks of K and V.
// grid = (B*G*NSEL, 4): block (x=bg*16+j, y=quarter) copies 16 rows x 128 f32
// from each of keys/values with float4 (GLOBAL_LOAD/STORE_B128) accesses.
// ---------------------------------------------------------------------------
__global__ __launch_bounds__(256) void nsa_gather_kernel(
    const float* __restrict__ keys,    // (B,G,SEQ,D)
    const float* __restrict__ values,  // (B,G,SEQ,D)
    const int* __restrict__ sel,       // (B*G*NSEL)
    float* __restrict__ out)           // [sel_keys | sel_values] flat
{
    const int bgj = blockIdx.x;        // 0..255
    const int j   = bgj & (NSEL - 1);  // selected slot (top-k rank order)
    const int bg  = bgj >> 4;          // b*G+g, 0..15
    const int q4  = blockIdx.y;        // 0..3 : which 16-row quarter

    const int blk = sel[bg * NSEL + j];
    const size_t src_row0 = (size_t)blk * BLK + (size_t)q4 * 16;
    const size_t dst_row0 = (size_t)j   * BLK + (size_t)q4 * 16;

    const float4* __restrict__ src_k =
        (const float4*)(keys   + ((size_t)bg * SEQ + src_row0) * D_DIM);
    const float4* __restrict__ src_v =
        (const float4*)(values + ((size_t)bg * SEQ + src_row0) * D_DIM);
    float4* __restrict__ dst_k =
        (float4*)(out             + ((size_t)bg * SEL_ROWS + dst_row0) * D_DIM);
    float4* __restrict__ dst_v =
        (float4*)(out + KEY_ELEMS + ((size_t)bg * SEL_ROWS + dst_row0) * D_DIM);

    // 16 rows * 32 float4/row = 512 float4 per tensor; 2 per thread.
    #pragma unroll
    for (int i = threadIdx.x; i < 16 * (D_DIM / 4); i += 256) {
        dst_k[i] = src_k[i];
        dst_v[i] = src_v[i];
    }
}

// ---------------------------------------------------------------------------
extern "C" void kernel_launch(void* const* d_in, const int* in_sizes, int n_in,
                              void* d_out, int out_size, void* d_ws, size_t ws_size,
                              hipStream_t stream) {
    const float* query  = (const float*)d_in[0]; // (4,32,1,128)
    const float* ck     = (const float*)d_in[1]; // (4,32,128,128)
    const float* keys   = (const float*)d_in[2]; // (4,4,8192,128)
    const float* values = (const float*)d_in[3]; // (4,4,8192,128)
    float* out = (float*)d_out;
    int*   sel = (int*)d_ws;                     // 256 ints of scratch

    nsa_select_kernel<<<B_DIM * G_DIM, 256, 0, stream>>>(query, ck, sel);

    dim3 grid(B_DIM * G_DIM * NSEL, 4);
    nsa_gather_kernel<<<grid, 256, 0, stream>>>(keys, values, sel, out);
}